// GCN_2Layers_10574209483123
// MI455X (gfx1250) — compile-verified
//
#include <hip/hip_runtime.h>

typedef __attribute__((ext_vector_type(2))) float v2f;
typedef __attribute__((ext_vector_type(8))) float v8f;

// ---------------------------------------------------------------------------
// Utility: zero a float buffer (grid-stride)
// ---------------------------------------------------------------------------
__global__ void zero_f32_kernel(float* __restrict__ p, long long n) {
    long long i = (long long)blockIdx.x * blockDim.x + threadIdx.x;
    long long stride = (long long)gridDim.x * blockDim.x;
    for (; i < n; i += stride) p[i] = 0.0f;
}

// ---------------------------------------------------------------------------
// Degree accumulation: deg_out[src[e]] += 1 ; deg_in[dst[e]] += 1
// ---------------------------------------------------------------------------
__global__ void degree_kernel(const int* __restrict__ src, const int* __restrict__ dst,
                              float* __restrict__ deg_out, float* __restrict__ deg_in,
                              int n_edges) {
    int i = blockIdx.x * blockDim.x + threadIdx.x;
    if (i < n_edges) {
        atomicAdd(&deg_out[src[i]], 1.0f);
        atomicAdd(&deg_in[dst[i]], 1.0f);
    }
}

// In-place: d[i] = rsqrt(max(d[i], 1))
__global__ void norm_kernel(float* __restrict__ d, long long n) {
    long long i = (long long)blockIdx.x * blockDim.x + threadIdx.x;
    if (i < n) d[i] = rsqrtf(fmaxf(d[i], 1.0f));
}

// ---------------------------------------------------------------------------
// WMMA GEMM:  Y[m, 0:64] = (X[m, 0:K] * norm[m]) @ W[K, 64]
// One wave (32 threads) per 16-row tile. nNodes must be a multiple of 16
// (100000 = 6250 * 16). Uses V_WMMA_F32_16X16X4_F32: exact fp32 math.
//
// A frag (16x4 f32, 2 VGPR): lane%16 = M row; lanes 0-15 hold K = k0+{0,1},
//                            lanes 16-31 hold K = k0+{2,3}.
// B frag (4x16 f32, 2 VGPR): lane%16 = N col; same K split across halves.
// C/D   (16x16 f32, 8 VGPR): VGPR r -> M = r (lanes 0-15) / r+8 (lanes 16-31),
//                            lane%16 = N col.
// ---------------------------------------------------------------------------
template <int K>
__global__ __launch_bounds__(32)
void gcn_gemm_wmma_kernel(const float* __restrict__ X, const float* __restrict__ W,
                          const float* __restrict__ norm, float* __restrict__ Y) {
    const int lane = threadIdx.x;          // 0..31
    const int half = lane >> 4;            // 0 or 1
    const int l16  = lane & 15;            // 0..15
    const long long rowA = (long long)blockIdx.x * 16 + l16;  // node row (A operand)
    const float s = norm[rowA];            // per-row source-degree normalization

    v8f acc0 = {}, acc1 = {}, acc2 = {}, acc3 = {};

    const float* xrow = X + rowA * K;
    for (int k0 = 0; k0 < K; k0 += 4) {
        // A fragment: two consecutive K values for this lane's half
        const float* ap = xrow + k0 + half * 2;
        v2f a;
        a.x = ap[0] * s;
        a.y = ap[1] * s;

        // B fragments for the 4 column tiles (W is [K,64] row-major)
        const float* wp = W + (long long)(k0 + half * 2) * 64;
        v2f b0, b1, b2, b3;
        b0.x = wp[l16 +  0];  b0.y = wp[64 + l16 +  0];
        b1.x = wp[l16 + 16];  b1.y = wp[64 + l16 + 16];
        b2.x = wp[l16 + 32];  b2.y = wp[64 + l16 + 32];
        b3.x = wp[l16 + 48];  b3.y = wp[64 + l16 + 48];

        acc0 = __builtin_amdgcn_wmma_f32_16x16x4_f32(false, a, false, b0, (short)0, acc0, false, false);
        acc1 = __builtin_amdgcn_wmma_f32_16x16x4_f32(false, a, false, b1, (short)0, acc1, false, false);
        acc2 = __builtin_amdgcn_wmma_f32_16x16x4_f32(false, a, false, b2, (short)0, acc2, false, false);
        acc3 = __builtin_amdgcn_wmma_f32_16x16x4_f32(false, a, false, b3, (short)0, acc3, false, false);
    }

    // Store D: row = tile_base + r + half*8, col = ntile*16 + l16
    float* yp = Y + ((long long)blockIdx.x * 16 + half * 8) * 64 + l16;
#pragma unroll
    for (int r = 0; r < 8; ++r) {
        yp[(long long)r * 64 +  0] = acc0[r];
        yp[(long long)r * 64 + 16] = acc1[r];
        yp[(long long)r * 64 + 32] = acc2[r];
        yp[(long long)r * 64 + 48] = acc3[r];
    }
}

// ---------------------------------------------------------------------------
// Edge scatter-add: out[dst[e], f] += H[src[e], f]   (64 features per edge)
// Thread t -> edge t/64, feature t%64 (coalesced feature access).
// H (25.6 MB) fits in the 192 MB L2, so gathers are L2 hits.
// ---------------------------------------------------------------------------
__global__ void scatter_add_kernel(const float* __restrict__ H, const int* __restrict__ src,
                                   const int* __restrict__ dst, float* __restrict__ out,
                                   long long total) {
    long long t = (long long)blockIdx.x * blockDim.x + threadIdx.x;
    if (t >= total) return;
    int e = (int)(t >> 6);
    int f = (int)(t & 63);
    float v = H[(long long)src[e] * 64 + f];
    atomicAdd(&out[(long long)dst[e] * 64 + f], v);
}

// ---------------------------------------------------------------------------
// Finalize: out[i,f] = out[i,f] * norm_dst[i] + bias[f]  (optional ReLU), in place
// ---------------------------------------------------------------------------
__global__ void finalize_kernel(float* __restrict__ out, const float* __restrict__ norm,
                                const float* __restrict__ bias, long long total, int relu) {
    long long t = (long long)blockIdx.x * blockDim.x + threadIdx.x;
    if (t >= total) return;
    long long i = t >> 6;
    int f = (int)(t & 63);
    float v = out[t] * norm[i] + bias[f];
    if (relu) v = fmaxf(v, 0.0f);
    out[t] = v;
}

// ---------------------------------------------------------------------------
// Host launcher
// ---------------------------------------------------------------------------
extern "C" void kernel_launch(void* const* d_in, const int* in_sizes, int n_in,
                              void* d_out, int out_size, void* d_ws, size_t ws_size,
                              hipStream_t stream) {
    const float* x  = (const float*)d_in[0];   // [N, 128]
    const int*   src = (const int*)d_in[1];    // [E]
    const int*   dst = (const int*)d_in[2];    // [E]
    const float* W1 = (const float*)d_in[3];   // [128, 64]
    const float* b1 = (const float*)d_in[4];   // [64]
    const float* W2 = (const float*)d_in[5];   // [64, 64]
    const float* b2 = (const float*)d_in[6];   // [64]

    const int nNodes = in_sizes[0] / 128;      // 100000 (multiple of 16)
    const int nEdges = in_sizes[1];            // 1600000

    float* out = (float*)d_out;
    float* h2 = out;                                // [N,64] first output
    float* h1 = out + (long long)nNodes * 64;       // [N,64] second output

    // Workspace: norm_src | norm_dst | projection buffer (reused by both layers)
    float* norm_src = (float*)d_ws;
    float* norm_dst = norm_src + nNodes;
    float* hproj    = norm_dst + nNodes;            // [N,64]

    const long long nOut   = (long long)nNodes * 64;
    const long long eTotal = (long long)nEdges * 64;
    const int TPB = 256;

    // 1) Zero degree buffers (2N) and both output regions (2*N*64).
    zero_f32_kernel<<<(unsigned)((2LL * nNodes + TPB - 1) / TPB), TPB, 0, stream>>>(norm_src, 2LL * nNodes);
    zero_f32_kernel<<<(unsigned)((2LL * nOut + TPB - 1) / TPB), TPB, 0, stream>>>(out, 2LL * nOut);

    // 2) Degrees -> 3) rsqrt(max(deg,1)) in place (both arrays contiguous).
    degree_kernel<<<(nEdges + TPB - 1) / TPB, TPB, 0, stream>>>(src, dst, norm_src, norm_dst, nEdges);
    norm_kernel<<<(unsigned)((2LL * nNodes + TPB - 1) / TPB), TPB, 0, stream>>>(norm_src, 2LL * nNodes);

    // ---- Layer 1 ----
    gcn_gemm_wmma_kernel<128><<<nNodes / 16, 32, 0, stream>>>(x, W1, norm_src, hproj);
    scatter_add_kernel<<<(unsigned)((eTotal + TPB - 1) / TPB), TPB, 0, stream>>>(hproj, src, dst, h1, eTotal);
    finalize_kernel<<<(unsigned)((nOut + TPB - 1) / TPB), TPB, 0, stream>>>(h1, norm_dst, b1, nOut, /*relu=*/1);

    // ---- Layer 2 (input = h1 in d_out, projection buffer reused) ----
    gcn_gemm_wmma_kernel<64><<<nNodes / 16, 32, 0, stream>>>(h1, W2, norm_src, hproj);
    scatter_add_kernel<<<(unsigned)((eTotal + TPB - 1) / TPB), TPB, 0, stream>>>(hproj, src, dst, h2, eTotal);
    finalize_kernel<<<(unsigned)((nOut + TPB - 1) / TPB), TPB, 0, stream>>>(h2, norm_dst, b2, nOut, /*relu=*/0);
}